// Encoder_8933531975861
// MI455X (gfx1250) — compile-verified
//
#include <hip/hip_runtime.h>
#include <math.h>

typedef __attribute__((ext_vector_type(2))) float v2f;
typedef __attribute__((ext_vector_type(8))) float v8f;
typedef __attribute__((ext_vector_type(4))) unsigned int v4u;
typedef __attribute__((ext_vector_type(8))) int v8i;
typedef __attribute__((ext_vector_type(4))) int v4i;

// ---------------- constants ----------------
#define N_IMG   509
#define N_C1    253          // (509-5)/2+1
#define N_C2    125          // (253-5)/2+1
#define C1PIX   64009        // 253*253
#define C2PIX   15625        // 125*125
#define NROW    32768        // 512*64
#define KDIM    192
#define NPAD    32           // 25 harmonics padded to 32
#define YPAD_DW 6144         // 192*32 dwords

static constexpr float INV_SQRT20  = 0.22360679774997896f;   // 1/sqrt(20)
static constexpr float INV_SQRT192 = 0.07216878364870322f;   // 1/sqrt(192)
static constexpr float INV_SQRT512 = 0.04419417382415922f;   // 1/sqrt(512)
static constexpr float SQRT2       = 1.4142135623730951f;
static constexpr float INV576      = 1.0f / 576.0f;

// ---------------- CNN featurizer (sample 0 only) ----------------
__global__ void k_conv1(const float* __restrict__ x, const float* __restrict__ w,
                        const float* __restrict__ b, float* __restrict__ out) {
  int idx = blockIdx.x * blockDim.x + threadIdx.x;
  if (idx >= 3 * C1PIX) return;
  int c = idx / C1PIX, r = idx % C1PIX;
  int i = r / N_C1, j = r % N_C1;
  float acc = b[c];
  const float* xp = x + (2 * i) * N_IMG + 2 * j;   // batch sample 0
  const float* wp = w + c * 25;
#pragma unroll
  for (int u = 0; u < 5; ++u)
#pragma unroll
    for (int v = 0; v < 5; ++v)
      acc += xp[u * N_IMG + v] * wp[u * 5 + v];
  out[idx] = fmaxf(acc, 0.0f);                      // fused ReLU
}

__global__ void k_conv2(const float* __restrict__ f1, const float* __restrict__ w,
                        const float* __restrict__ b, float* __restrict__ out) {
  int idx = blockIdx.x * blockDim.x + threadIdx.x;
  if (idx >= 5 * C2PIX) return;
  int o = idx / C2PIX, r = idx % C2PIX;
  int i = r / N_C2, j = r % N_C2;
  float acc = b[o];
#pragma unroll
  for (int c = 0; c < 3; ++c) {
    const float* fp = f1 + c * C1PIX + (2 * i) * N_C1 + 2 * j;
    const float* wp = w + (o * 3 + c) * 25;
#pragma unroll
    for (int u = 0; u < 5; ++u)
#pragma unroll
      for (int v = 0; v < 5; ++v)
        acc += fp[u * N_C1 + v] * wp[u * 5 + v];
  }
  out[idx] = acc;
}

// ---------------- projector: fp[c,p] = sum_hw f0[c,hw]*projw[hw,p]; row c=5 is sum(projw) ----------------
__global__ void k_fp(const float* __restrict__ f0, const float* __restrict__ pw,
                     float* __restrict__ fp) {
  __shared__ float red[256];
  int c = blockIdx.x / 20, p = blockIdx.x % 20;
  float acc = 0.0f;
  for (int i = threadIdx.x; i < C2PIX; i += 256) {
    float v = (c < 5) ? f0[c * C2PIX + i] : 1.0f;
    acc += v * pw[i * 20 + p];
  }
  red[threadIdx.x] = acc;
  __syncthreads();
  for (int s = 128; s > 0; s >>= 1) {
    if (threadIdx.x < s) red[threadIdx.x] += red[threadIdx.x + s];
    __syncthreads();
  }
  if (threadIdx.x == 0) fp[c * 20 + p] = red[0];
}

__global__ void k_pts(const float* __restrict__ fp, const float* __restrict__ p_w,
                      const float* __restrict__ p_b, float* __restrict__ pts) {
  int idx = blockIdx.x * blockDim.x + threadIdx.x;   // 512*20
  if (idx >= 512 * 20) return;
  int o = idx / 20, p = idx % 20;
  float acc = p_b[o] * fp[5 * 20 + p];               // bias * sum(projw col p)
#pragma unroll
  for (int c = 0; c < 5; ++c) acc += p_w[o * 5 + c] * fp[c * 20 + p];
  pts[idx] = acc;
}

__global__ void k_harm(const float* __restrict__ pts, const float* __restrict__ Yproj,
                       float* __restrict__ harm) {
  int idx = blockIdx.x * blockDim.x + threadIdx.x;   // 512*25
  if (idx >= 512 * 25) return;
  int o = idx / 25, m = idx % 25;
  float acc = 0.0f;
#pragma unroll
  for (int p = 0; p < 20; ++p) acc += pts[o * 20 + p] * Yproj[p * 25 + m];
  harm[idx] = acc * INV_SQRT20;
}

// ---------------- pad Y_s2 (192x25) -> K-pair-interleaved (96,32,2), fold 1/sqrt(192) ----------------
// Ypad[(k>>1)*64 + 2*n + (k&1)] = Ys2[k,n]/sqrt(192)  (zero for n>=25)
// so a lane's B operand (rows K, K+1 at column n) is one contiguous, 8B-aligned float2.
__global__ void k_padY(const float* __restrict__ Ys2, float* __restrict__ Ypad) {
  int t = blockIdx.x * blockDim.x + threadIdx.x;     // 192*32
  if (t >= KDIM * NPAD) return;
  int k = t / NPAD, n = t % NPAD;
  float v = (n < 25) ? Ys2[k * 25 + n] * INV_SQRT192 : 0.0f;
  Ypad[(k >> 1) * 64 + 2 * n + (k & 1)] = v;
}

// ---------------- psi = s2_w(32768x192) @ Y(192x32), f32 WMMA 16x16x4 ----------------
// B staged in LDS via Tensor Data Mover (one tensor_load_to_lds per workgroup).
// Output stored y-major: psi[(y*512 + x)*32 + col] so stage B streams coalesced.
__global__ __launch_bounds__(256) void k_psi_gemm(const float* __restrict__ W,
                                                  const float* __restrict__ Ypad,
                                                  float* __restrict__ psi) {
  __shared__ float ldsY[YPAD_DW];                    // 24 KB, K-pair-interleaved layout

  // ---- TDM: stage Ypad (6144 dwords, 1-D tile) into LDS; wave 0 issues ----
  if (threadIdx.x < 32) {
    unsigned long long ga = (unsigned long long)(size_t)Ypad;
    unsigned lds_off = (unsigned)(size_t)(&ldsY[0]);
    v4u g0;
    g0[0] = 1u;                                      // count=1 (valid), user mode
    g0[1] = lds_off;                                 // lds_addr (bytes)
    g0[2] = (unsigned)ga;                            // global_addr[31:0]
    g0[3] = (unsigned)((ga >> 32) & 0x1FFFFFFull) | (2u << 30);  // addr[56:32] | type=2
    v8i g1;
    g1[0] = 0x00020000;                              // workgroup_mask=0, data_size=2 (4B)
    g1[1] = (int)(6144u << 16);                      // tensor_dim0 = 6144 (low 16 in [31:16])
    g1[2] = 0x00010000;                              // tensor_dim0 hi=0, tensor_dim1=1
    g1[3] = (int)(6144u << 16);                      // tile_dim0 = 6144
    g1[4] = 0;                                       // tile_dim1=0, tile_dim2=0 (unused)
    g1[5] = 6144;                                    // tensor_dim0_stride lo
    g1[6] = 0;
    g1[7] = 0;
    v4i g2 = {0, 0, 0, 0};
    v4i g3 = {0, 0, 0, 0};
    v8i g4 = {0, 0, 0, 0, 0, 0, 0, 0};
    __builtin_amdgcn_tensor_load_to_lds(g0, g1, g2, g3, g4, 0);
    __builtin_amdgcn_s_wait_tensorcnt(0);
  }
  __syncthreads();

  const int lane = threadIdx.x & 31;
  const int wave = threadIdx.x >> 5;                 // 0..7
  const int m0   = (blockIdx.x * 8 + wave) * 16;     // M tile base (0..32752), within one x
  const int r    = lane & 15;                        // A: row in tile / B,D: column
  const int kb   = (lane >> 4) << 1;                 // 0 or 2 per lane-half
  v8f c0 = {};
  v8f c1 = {};
  const float* wrow = W + (size_t)(m0 + r) * KDIM;
  const float* ypair = ldsY + (kb >> 1) * 64;        // K-pair base for this lane-half
  for (int k = 0; k < KDIM; k += 4) {
    // A (16x4): lane=M, v0 holds K=kb, v1 holds K=kb+1
    v2f a = *(const v2f*)(wrow + k + kb);
    // B (4x16): rows (K=k+kb, K=k+kb+1) at column n -> one contiguous float2 in LDS
    v2f b0 = *(const v2f*)(ypair + (k >> 1) * 64 + 2 * r);
    v2f b1 = *(const v2f*)(ypair + (k >> 1) * 64 + 2 * (16 + r));
    c0 = __builtin_amdgcn_wmma_f32_16x16x4_f32(false, a, false, b0, (short)0, c0, false, false);
    c1 = __builtin_amdgcn_wmma_f32_16x16x4_f32(false, a, false, b1, (short)0, c1, false, false);
  }
  // D (16x16 f32): VGPR j -> M = j + 8*(lane/16), N = lane%16
  // W row index m = x*64 + y; tiles never cross an x boundary (m0 % 64 in {0,16,32,48}).
  const int x  = m0 >> 6;
  const int y0 = (m0 & 63) + ((lane >> 4) * 8);
#pragma unroll
  for (int j = 0; j < 8; ++j) {
    size_t row = (size_t)(y0 + j) * 512 + x;         // y-major
    psi[row * NPAD + r]      = c0[j];
    psi[row * NPAD + 16 + r] = c1[j];
  }
}

// ---------------- S2Conv block contraction: h[y, t(l,u,m)] = sum_x psi[y,x,l2+u]*harm[x,l2+m]/sqrt(512) ----------------
__global__ void k_h(const float* __restrict__ psi, const float* __restrict__ harm,
                    float* __restrict__ h) {
  int y = blockIdx.x;
  int t = threadIdx.x;
  if (t >= 165) return;
  int l = 0, rem = t;
  while (rem >= (2 * l + 1) * (2 * l + 1)) { rem -= (2 * l + 1) * (2 * l + 1); ++l; }
  int d = 2 * l + 1;
  int u = rem / d, m = rem % d;
  int ho = l * l;                                    // harmonic offset
  const int colA = ho + u, colB = ho + m;
  const float* pbase = psi + (size_t)y * 512 * NPAD; // y-major layout
  float acc = 0.0f;
  for (int x = 0; x < 512; ++x)
    acc += pbase[(size_t)x * NPAD + colA] * harm[x * 25 + colB];
  h[y * 165 + t] = acc * INV_SQRT512;
}

// ---------------- SO3 activation ----------------
__global__ void k_act1(const float* __restrict__ h, const float* __restrict__ Dact,
                       float* __restrict__ s) {
  int idx = blockIdx.x * blockDim.x + threadIdx.x;   // 64*576
  if (idx >= 64 * 576) return;
  int f = idx / 576, g = idx % 576;
  float acc = 0.0f;
  const float* hr = h + f * 165;
  const float* dr = Dact + g * 165;
  for (int i = 0; i < 165; ++i) acc += hr[i] * dr[i];
  s[idx] = SQRT2 * fmaxf(acc, 0.0f);
}

__global__ void k_act2(const float* __restrict__ s, const float* __restrict__ Dact,
                       float* __restrict__ h2) {
  int idx = blockIdx.x * blockDim.x + threadIdx.x;   // 64*165
  if (idx >= 64 * 165) return;
  int f = idx / 165, i = idx % 165;
  float acc = 0.0f;
  const float* sr = s + f * 576;
  for (int g = 0; g < 576; ++g) acc += sr[g] * Dact[g * 165 + i];
  h2[idx] = acc * INV576;
}

// ---------------- BatchNorm over the 64 'channel' rows (training stats), in place ----------------
__global__ void k_bn(float* __restrict__ h2, const float* __restrict__ gam,
                     const float* __restrict__ bet) {
  int i = threadIdx.x;
  if (i >= 165) return;
  float mu = 0.0f;
  for (int f = 0; f < 64; ++f) mu += h2[f * 165 + i];
  mu *= (1.0f / 64.0f);
  float var = 0.0f;
  for (int f = 0; f < 64; ++f) { float d = h2[f * 165 + i] - mu; var += d * d; }
  var *= (1.0f / 64.0f);
  float sc = rsqrtf(var + 1e-5f) * gam[i];
  float sh = bet[i];
  for (int f = 0; f < 64; ++f) h2[f * 165 + i] = (h2[f * 165 + i] - mu) * sc + sh;
}

// ---------------- SO3Conv kernel expansion: psi2[y,i] = sum_n D_so3k[n,i]*so3_w[y,n]/sqrt(192) ----------------
__global__ void k_psi2(const float* __restrict__ Dso3k, const float* __restrict__ so3w,
                       float* __restrict__ psi2) {
  int idx = blockIdx.x * blockDim.x + threadIdx.x;   // 64*165
  if (idx >= 64 * 165) return;
  int y = idx / 165, i = idx % 165;
  float acc = 0.0f;
  const float* wr = so3w + y * 192;
  for (int n = 0; n < 192; ++n) acc += Dso3k[n * 165 + i] * wr[n];
  psi2[idx] = acc * INV_SQRT192;
}

// ---------------- SO3Conv contraction: harm_out[t(l,u,m)] = sum_y sum_v psi2[y,off+u*d+v]*h2[y,off+v*d+m]/sqrt(64d) ----------------
__global__ void k_harmout(const float* __restrict__ psi2, const float* __restrict__ h2,
                          float* __restrict__ hout) {
  int t = threadIdx.x;
  if (t >= 165) return;
  int l = 0, rem = t;
  while (rem >= (2 * l + 1) * (2 * l + 1)) { rem -= (2 * l + 1) * (2 * l + 1); ++l; }
  int d = 2 * l + 1;
  int u = rem / d, m = rem % d;
  int off = t - rem;                                 // start of l's d*d block in 165-flat
  float acc = 0.0f;
  for (int y = 0; y < 64; ++y) {
    const float* pr = psi2 + y * 165 + off + u * d;
    const float* hr = h2 + y * 165 + off + m;
    for (int v = 0; v < d; ++v) acc += pr[v] * hr[v * d];
  }
  hout[t] = acc * rsqrtf(64.0f * (float)d);
}

// ---------------- final: sigmoid(harm_out @ eval_wigners) ----------------
__global__ void k_final(const float* __restrict__ hout, const float* __restrict__ evw,
                        float* __restrict__ out) {
  int g = blockIdx.x * blockDim.x + threadIdx.x;     // 4608
  if (g >= 4608) return;
  float acc = 0.0f;
  for (int i = 0; i < 165; ++i) acc += hout[i] * evw[i * 4608 + g];
  out[g] = 1.0f / (1.0f + expf(-acc));
}

// ---------------- host launcher ----------------
extern "C" void kernel_launch(void* const* d_in, const int* in_sizes, int n_in,
                              void* d_out, int out_size, void* d_ws, size_t ws_size,
                              hipStream_t stream) {
  (void)in_sizes; (void)n_in; (void)out_size; (void)ws_size;
  const float* x     = (const float*)d_in[0];
  const float* w1    = (const float*)d_in[1];
  const float* b1    = (const float*)d_in[2];
  const float* w2    = (const float*)d_in[3];
  const float* b2    = (const float*)d_in[4];
  const float* p_w   = (const float*)d_in[5];
  const float* p_b   = (const float*)d_in[6];
  const float* projw = (const float*)d_in[7];
  const float* Yproj = (const float*)d_in[8];
  const float* s2w   = (const float*)d_in[9];
  const float* Ys2   = (const float*)d_in[10];
  const float* gam   = (const float*)d_in[11];
  const float* bet   = (const float*)d_in[12];
  const float* so3w  = (const float*)d_in[13];
  const float* Dso3k = (const float*)d_in[14];
  const float* Dact  = (const float*)d_in[15];
  const float* evw   = (const float*)d_in[16];
  float* out = (float*)d_out;

  float* ws    = (float*)d_ws;
  float* feat1 = ws; ws += 192032;        // (3,253,253)
  float* f0    = ws; ws += 78128;         // (5,125,125)
  float* fp    = ws; ws += 128;           // (6,20)
  float* pts   = ws; ws += 10240;         // (512,20)
  float* harm  = ws; ws += 12800;         // (512,25)
  float* Ypad  = ws; ws += YPAD_DW;       // (96,32,2) K-pair interleaved
  float* psi   = ws; ws += (size_t)NROW * NPAD; // (64,512,32) y-major, ~4MB
  float* h     = ws; ws += 10560;         // (64,165)
  float* s     = ws; ws += 36864;         // (64,576)
  float* h2    = ws; ws += 10560;         // (64,165)
  float* psi2  = ws; ws += 10560;         // (64,165)
  float* hout  = ws; ws += 192;           // (165)

  k_conv1<<<(3 * C1PIX + 255) / 256, 256, 0, stream>>>(x, w1, b1, feat1);
  k_conv2<<<(5 * C2PIX + 255) / 256, 256, 0, stream>>>(feat1, w2, b2, f0);
  k_fp<<<120, 256, 0, stream>>>(f0, projw, fp);
  k_pts<<<40, 256, 0, stream>>>(fp, p_w, p_b, pts);
  k_harm<<<50, 256, 0, stream>>>(pts, Yproj, harm);
  k_padY<<<24, 256, 0, stream>>>(Ys2, Ypad);
  k_psi_gemm<<<256, 256, 0, stream>>>(s2w, Ypad, psi);
  k_h<<<64, 192, 0, stream>>>(psi, harm, h);
  k_act1<<<144, 256, 0, stream>>>(h, Dact, s);
  k_act2<<<(64 * 165 + 255) / 256, 256, 0, stream>>>(s, Dact, h2);
  k_bn<<<1, 192, 0, stream>>>(h2, gam, bet);
  k_psi2<<<(64 * 165 + 255) / 256, 256, 0, stream>>>(Dso3k, so3w, psi2);
  k_harmout<<<1, 192, 0, stream>>>(psi2, h2, hout);
  k_final<<<(4608 + 255) / 256, 256, 0, stream>>>(hout, evw, out);
}